// DenoisingConvNN_2D_K_N_Location_25039659335750
// MI455X (gfx1250) — compile-verified
//
#include <hip/hip_runtime.h>
#include <math.h>
#include <stdint.h>

typedef __attribute__((ext_vector_type(2))) float v2f;
typedef __attribute__((ext_vector_type(8))) float v8f;

#define BATCH 16
#define TPIX  4096   // 64*64
#define NSAMP 64
#define KNBR  9
#define EPSV  1e-8f

// ---------------------------------------------------------------------------
// Build layer-1 xf: pixel_unshuffle(x,2) -> channels 0..11 of xf (B,16,T)
// ---------------------------------------------------------------------------
__global__ void build_xf1_kernel(const float* __restrict__ x, float* __restrict__ xf,
                                 int Cp) {
    long g = (long)blockIdx.x * blockDim.x + threadIdx.x;
    long total = (long)BATCH * 12 * TPIX;
    if (g >= total) return;
    int t  = (int)(g % TPIX);
    int cu = (int)((g / TPIX) % 12);
    int b  = (int)(g / (12L * TPIX));
    int c = cu >> 2, sy = (cu >> 1) & 1, sx = cu & 1;
    int hs = t >> 6, ws = t & 63;
    xf[((size_t)b * Cp + cu) * TPIX + t] =
        x[(((size_t)b * 3 + c) * 128 + (hs * 2 + sy)) * 128 + (ws * 2 + sx)];
}

// ---------------------------------------------------------------------------
// Fill coordinate channels (C-2 = gy, C-1 = gx) and zero the pad channels up
// to Cp. gy/gx = linspace(-1,1,64).
// ---------------------------------------------------------------------------
__global__ void fill_coords_kernel(float* __restrict__ xf, int C, int Cp) {
    int nch = Cp - (C - 2);
    long total = (long)BATCH * nch * TPIX;
    long g = (long)blockIdx.x * blockDim.x + threadIdx.x;
    if (g >= total) return;
    int t  = (int)(g % TPIX);
    int ci = (int)((g / TPIX) % nch);
    int b  = (int)(g / ((long)TPIX * nch));
    int ch = (C - 2) + ci;
    int hs = t >> 6, ws = t & 63;
    float v = 0.0f;
    if (ch == C - 2)      v = -1.0f + 2.0f * (float)hs / 63.0f;
    else if (ch == C - 1) v = -1.0f + 2.0f * (float)ws / 63.0f;
    xf[((size_t)b * Cp + ch) * TPIX + t] = v;
}

// ---------------------------------------------------------------------------
// Sample columns: xs[b,c,n] = xf[b,c,idx[n]] ; xsn = xs / (||xs||_c + eps)
// One thread per (b,n).
// ---------------------------------------------------------------------------
__global__ void sample_kernel(const float* __restrict__ xf, const int* __restrict__ idx,
                              float* __restrict__ xs, float* __restrict__ xsn, int Cp) {
    int g = blockIdx.x * blockDim.x + threadIdx.x;
    if (g >= BATCH * NSAMP) return;
    int n = g % NSAMP, b = g / NSAMP;
    int t = idx[n];
    float ss = 0.0f;
    for (int c = 0; c < Cp; ++c) {
        float v = xf[((size_t)b * Cp + c) * TPIX + t];
        xs[((size_t)b * Cp + c) * NSAMP + n] = v;
        ss += v * v;
    }
    float inv = 1.0f / (sqrtf(ss) + EPSV);
    for (int c = 0; c < Cp; ++c)
        xsn[((size_t)b * Cp + c) * NSAMP + n] =
            xs[((size_t)b * Cp + c) * NSAMP + n] * inv;
}

// ---------------------------------------------------------------------------
// sim[b,t,n] = sum_c xf[b,c,t] * xsn[b,c,n]   (row scale dropped: does not
// affect per-row top-k). V_WMMA_F32_16X16X4_F32, one wave per 16x16 tile.
// ---------------------------------------------------------------------------
__global__ void sim_wmma_kernel(const float* __restrict__ xf,
                                const float* __restrict__ xsn,
                                float* __restrict__ sim, int Cp) {
    const int lane = threadIdx.x & 31;
    const int wave = threadIdx.x >> 5;
    const int tilesT = TPIX / 16, tilesN = NSAMP / 16;
    long tile = (long)blockIdx.x * (blockDim.x >> 5) + wave;   // wave-uniform
    if (tile >= (long)BATCH * tilesT * tilesN) return;         // uniform branch

    int nt = (int)(tile % tilesN);
    int tt = (int)((tile / tilesN) % tilesT);
    int b  = (int)(tile / ((long)tilesN * tilesT));
    int t0 = tt * 16, n0 = nt * 16;

    const float* xfb = xf  + (size_t)b * Cp * TPIX;
    const float* xsb = xsn + (size_t)b * Cp * NSAMP;

    const int half  = lane >> 4;      // 0: K=0,1   1: K=2,3
    const int l16   = lane & 15;      // A row / B col
    const int kbase = half * 2;

    v8f acc = {};
    for (int c0 = 0; c0 < Cp; c0 += 4) {
        v2f a, bm;
        a.x  = xfb[(size_t)(c0 + kbase + 0) * TPIX + (t0 + l16)];
        a.y  = xfb[(size_t)(c0 + kbase + 1) * TPIX + (t0 + l16)];
        bm.x = xsb[(size_t)(c0 + kbase + 0) * NSAMP + (n0 + l16)];
        bm.y = xsb[(size_t)(c0 + kbase + 1) * NSAMP + (n0 + l16)];
        acc = __builtin_amdgcn_wmma_f32_16x16x4_f32(false, a, false, bm,
                                                    (short)0, acc, false, false);
    }
    // C/D layout: VGPR j -> (M=j, N=lane) lanes 0-15; (M=j+8, N=lane-16) lanes 16-31
    float* sb = sim + ((size_t)b * TPIX + t0) * NSAMP + n0;
#pragma unroll
    for (int j = 0; j < 8; ++j) {
        int M = j + half * 8;
        sb[(size_t)M * NSAMP + l16] = acc[j];
    }
}

// ---------------------------------------------------------------------------
// Stable top-9 of 64 per (b,t) using a 64-bit exclusion mask (ties -> lowest
// index first, matching jax.lax.top_k).
// ---------------------------------------------------------------------------
__global__ void topk_kernel(const float* __restrict__ sim, int* __restrict__ nbr) {
    long g = (long)blockIdx.x * blockDim.x + threadIdx.x;
    if (g >= (long)BATCH * TPIX) return;
    const float* s = sim + (size_t)g * NSAMP;
    int* out = nbr + (size_t)g * KNBR;
    uint64_t used = 0;
    for (int kk = 0; kk < KNBR; ++kk) {
        float best = -__builtin_inff();
        int bi = 0;
        for (int n = 0; n < NSAMP; ++n) {
            float v = s[n];
            bool ok = ((used >> n) & 1ull) == 0ull;
            if (ok && v > best) { best = v; bi = n; }
        }
        out[kk] = bi;
        used |= 1ull << bi;
    }
}

// ---------------------------------------------------------------------------
// y[b,k,o,n] = sum_c w[o,c,k] * xs[b,c,n]  — small GEMM per (b,k), WMMA f32.
// ---------------------------------------------------------------------------
__global__ void y_wmma_kernel(const float* __restrict__ w, const float* __restrict__ xs,
                              float* __restrict__ y, int O, int C, int Cp, int Op) {
    const int lane = threadIdx.x & 31;
    const int wave = threadIdx.x >> 5;
    const int tilesO = Op / 16, tilesN = NSAMP / 16;
    long tile = (long)blockIdx.x * (blockDim.x >> 5) + wave;   // wave-uniform
    if (tile >= (long)BATCH * KNBR * tilesO * tilesN) return;

    int nt = (int)(tile % tilesN);            long r = tile / tilesN;
    int ot = (int)(r % tilesO);               r /= tilesO;
    int k  = (int)(r % KNBR);
    int b  = (int)(r / KNBR);
    int o0 = ot * 16, n0 = nt * 16;

    const float* xsb = xs + (size_t)b * Cp * NSAMP;
    const int half  = lane >> 4;
    const int l16   = lane & 15;
    const int kbase = half * 2;
    const int o     = o0 + l16;

    v8f acc = {};
    for (int c0 = 0; c0 < Cp; c0 += 4) {
        int ca = c0 + kbase;
        v2f a, bm;
        a.x  = (o < O && (ca + 0) < C) ? w[((size_t)o * C + (ca + 0)) * KNBR + k] : 0.0f;
        a.y  = (o < O && (ca + 1) < C) ? w[((size_t)o * C + (ca + 1)) * KNBR + k] : 0.0f;
        bm.x = xsb[(size_t)(ca + 0) * NSAMP + (n0 + l16)];
        bm.y = xsb[(size_t)(ca + 1) * NSAMP + (n0 + l16)];
        acc = __builtin_amdgcn_wmma_f32_16x16x4_f32(false, a, false, bm,
                                                    (short)0, acc, false, false);
    }
    float* yb = y + (((size_t)b * KNBR + k) * Op + o0) * NSAMP + n0;
#pragma unroll
    for (int j = 0; j < 8; ++j) {
        int M = j + half * 8;
        yb[(size_t)M * NSAMP + l16] = acc[j];
    }
}

// ---------------------------------------------------------------------------
// out[o,t] = bias[o] + sum_k y[b,k,o,nbr[t,k]]  (+ ReLU). Writes either the
// next layer's xf channels 0..O-1 (pixel_shuffle∘unshuffle == identity), or
// the final (B,3,128,128) output via pixel_shuffle.
// ---------------------------------------------------------------------------
__global__ void gather_out_kernel(const float* __restrict__ y, const float* __restrict__ bias,
                                  const int* __restrict__ nbr, float* __restrict__ out,
                                  int O, int Op, int relu, int final_shuffle, int Cp_next) {
    long g = (long)blockIdx.x * blockDim.x + threadIdx.x;
    long total = (long)BATCH * TPIX * O;
    if (g >= total) return;
    int o = (int)(g % O);
    int t = (int)((g / O) % TPIX);
    int b = (int)(g / ((long)O * TPIX));

    const int*   nb = nbr + ((size_t)b * TPIX + t) * KNBR;
    const float* yb = y + (size_t)b * KNBR * Op * NSAMP;
    float acc = bias[o];
#pragma unroll
    for (int k = 0; k < KNBR; ++k) {
        int n = nb[k];
        acc += yb[((size_t)k * Op + o) * NSAMP + n];
    }
    if (relu) acc = fmaxf(acc, 0.0f);
    if (final_shuffle) {
        int c = o >> 2, sy = (o >> 1) & 1, sx = o & 1;
        int hs = t >> 6, ws = t & 63;
        out[(((size_t)b * 3 + c) * 128 + (hs * 2 + sy)) * 128 + (ws * 2 + sx)] = acc;
    } else {
        out[((size_t)b * Cp_next + o) * TPIX + t] = acc;
    }
}

// ---------------------------------------------------------------------------
// Host-side layer driver
// ---------------------------------------------------------------------------
static void run_layer(const float* xf_in, float* out,
                      const float* w, const float* bias, const int* idx,
                      int C, int Cp, int O, int Op, int relu, int final_mode, int Cp_next,
                      float* xs, float* xsn, float* simbuf, int* nbr, float* ybuf,
                      hipStream_t stream) {
    sample_kernel<<<(BATCH * NSAMP + 255) / 256, 256, 0, stream>>>(xf_in, idx, xs, xsn, Cp);

    long simTiles = (long)BATCH * (TPIX / 16) * (NSAMP / 16);
    sim_wmma_kernel<<<(int)((simTiles + 7) / 8), 256, 0, stream>>>(xf_in, xsn, simbuf, Cp);

    topk_kernel<<<(BATCH * TPIX + 255) / 256, 256, 0, stream>>>(simbuf, nbr);

    long yTiles = (long)BATCH * KNBR * (Op / 16) * (NSAMP / 16);
    y_wmma_kernel<<<(int)((yTiles + 7) / 8), 256, 0, stream>>>(w, xs, ybuf, O, C, Cp, Op);

    long total = (long)BATCH * TPIX * O;
    gather_out_kernel<<<(int)((total + 255) / 256), 256, 0, stream>>>(
        ybuf, bias, nbr, out, O, Op, relu, final_mode, Cp_next);
}

extern "C" void kernel_launch(void* const* d_in, const int* in_sizes, int n_in,
                              void* d_out, int out_size, void* d_ws, size_t ws_size,
                              hipStream_t stream) {
    const float* x    = (const float*)d_in[0];
    const float* w1   = (const float*)d_in[1];
    const float* b1   = (const float*)d_in[2];
    const float* w2   = (const float*)d_in[3];
    const float* b2   = (const float*)d_in[4];
    const float* w3   = (const float*)d_in[5];
    const float* b3   = (const float*)d_in[6];
    const int*   idx1 = (const int*)d_in[7];
    const int*   idx2 = (const int*)d_in[8];
    const int*   idx3 = (const int*)d_in[9];
    float* out = (float*)d_out;

    // Workspace carve (~94 MB total)
    const int CPMAX = 132, OPMAX = 128;
    size_t off = 0;
    char* ws = (char*)d_ws;
    auto carve = [&](size_t bytes) -> void* {
        off = (off + 255) & ~(size_t)255;
        void* p = ws + off;
        off += bytes;
        return p;
    };
    float* xfA    = (float*)carve((size_t)BATCH * CPMAX * TPIX * sizeof(float));
    float* xfB    = (float*)carve((size_t)BATCH * CPMAX * TPIX * sizeof(float));
    float* xs     = (float*)carve((size_t)BATCH * CPMAX * NSAMP * sizeof(float));
    float* xsn    = (float*)carve((size_t)BATCH * CPMAX * NSAMP * sizeof(float));
    float* simbuf = (float*)carve((size_t)BATCH * TPIX * NSAMP * sizeof(float));
    int*   nbr    = (int*)  carve((size_t)BATCH * TPIX * KNBR * sizeof(int));
    float* ybuf   = (float*)carve((size_t)BATCH * KNBR * OPMAX * NSAMP * sizeof(float));
    (void)ws_size; (void)in_sizes; (void)n_in; (void)out_size;

    // ---- Layer 1: C=14 (pad 16), O=64 ----
    build_xf1_kernel<<<(int)(((long)BATCH * 12 * TPIX + 255) / 256), 256, 0, stream>>>(x, xfA, 16);
    fill_coords_kernel<<<(int)(((long)BATCH * 4 * TPIX + 255) / 256), 256, 0, stream>>>(xfA, 14, 16);
    run_layer(xfA, xfB, w1, b1, idx1, /*C=*/14, /*Cp=*/16, /*O=*/64, /*Op=*/64,
              /*relu=*/1, /*final=*/0, /*Cp_next=*/68, xs, xsn, simbuf, nbr, ybuf, stream);

    // ---- Layer 2: C=66 (pad 68), O=128 ----
    fill_coords_kernel<<<(int)(((long)BATCH * 4 * TPIX + 255) / 256), 256, 0, stream>>>(xfB, 66, 68);
    run_layer(xfB, xfA, w2, b2, idx2, /*C=*/66, /*Cp=*/68, /*O=*/128, /*Op=*/128,
              /*relu=*/1, /*final=*/0, /*Cp_next=*/132, xs, xsn, simbuf, nbr, ybuf, stream);

    // ---- Layer 3: C=130 (pad 132), O=12 (pad 16), pixel_shuffle to output ----
    fill_coords_kernel<<<(int)(((long)BATCH * 4 * TPIX + 255) / 256), 256, 0, stream>>>(xfA, 130, 132);
    run_layer(xfA, out, w3, b3, idx3, /*C=*/130, /*Cp=*/132, /*O=*/12, /*Op=*/16,
              /*relu=*/0, /*final=*/1, /*Cp_next=*/0, xs, xsn, simbuf, nbr, ybuf, stream);
}